// RGNNLayer_26027501814526
// MI455X (gfx1250) — compile-verified
//
#include <hip/hip_runtime.h>

#define NN 100000
#define NE 1250000
#define DD 64
#define NR 4

typedef __attribute__((ext_vector_type(2))) float v2f;
typedef __attribute__((ext_vector_type(8))) float v8f;

// ---------------------------------------------------------------------------
// Kernel 0: zero the per-relation aggregation buffers in workspace.
// ---------------------------------------------------------------------------
__global__ __launch_bounds__(256) void zero_ws_kernel(float4* __restrict__ p, int n4) {
  int i = blockIdx.x * blockDim.x + threadIdx.x;
  int stride = gridDim.x * blockDim.x;
  float4 z = make_float4(0.f, 0.f, 0.f, 0.f);
  for (; i < n4; i += stride) p[i] = z;
}

// ---------------------------------------------------------------------------
// Kernel 1: scatter raw x rows into per-relation aggregation buffers.
//   agg_r[dst] += x[src]   (atomic f32 adds; lane l handles columns 2l,2l+1)
// One edge per wave per loop iteration; edge indices are wave-uniform.
// ---------------------------------------------------------------------------
__global__ __launch_bounds__(256) void scatter_agg_kernel(
    const float* __restrict__ x,
    const int* __restrict__ e0, const int* __restrict__ e1,
    const int* __restrict__ e2, const int* __restrict__ e3,
    float* __restrict__ agg) {
  const int rel = blockIdx.y;
  const int* e = (rel == 0) ? e0 : (rel == 1) ? e1 : (rel == 2) ? e2 : e3;
  float* aggr = agg + (size_t)rel * NN * DD;

  const int lane = threadIdx.x & 31;
  const int wid  = threadIdx.x >> 5;
  const int wavesPerBlock = blockDim.x >> 5;
  int wave   = blockIdx.x * wavesPerBlock + wid;
  int nwaves = gridDim.x * wavesPerBlock;

  for (int eIdx = wave; eIdx < NE; eIdx += nwaves) {
    int src = e[eIdx];        // edge_index[0][e]
    int dst = e[NE + eIdx];   // edge_index[1][e]
    const float2 v = *(const float2*)(x + (size_t)src * DD + lane * 2);
    float* dp = aggr + (size_t)dst * DD + lane * 2;
    atomicAdd(dp + 0, v.x);   // global_atomic_add_f32
    atomicAdd(dp + 1, v.y);
  }
}

// ---------------------------------------------------------------------------
// Kernel 2: fused WMMA GEMM pass.
//   out = x @ root_w^T + root_b + sum_r agg_r @ conv_w[r]^T
// One wave computes a 16x64 row-block; 4 N-tiles accumulated across 5 source
// matrices with V_WMMA_F32_16X16X4_F32 (16 K-steps x 4 N x 5 src = 320 WMMA).
//
// A 16x4 f32 layout: lane(0..15)->M, lane>>4 selects K pair; VGPR j -> K+j.
// B 4x16  f32 layout: lane&15 -> N col, lane>>4 selects K pair; VGPR j -> K+j.
// C/D 16x16 f32: VGPR v -> M = v + 8*(lane>>4), N = lane&15.
// ---------------------------------------------------------------------------
__global__ __launch_bounds__(256) void rgcn_wmma_gemm_kernel(
    const float* __restrict__ x,
    const float* __restrict__ root_w,
    const float* __restrict__ root_b,
    const float* __restrict__ conv_w,
    const float* __restrict__ agg,
    float* __restrict__ out) {
  const int lane  = threadIdx.x & 31;
  const int wid   = threadIdx.x >> 5;
  const int mtile = blockIdx.x * (blockDim.x >> 5) + wid;
  if (mtile >= NN / 16) return;   // whole wave exits together; EXEC stays all-1

  const int m0   = mtile * 16;
  const int arow = m0 + (lane & 15);   // A-matrix row this lane feeds
  const int koff = (lane >> 4) * 2;    // K sub-offset within a K=4 step
  const int ncol = lane & 15;          // B/C/D column within an N-tile

  v8f acc[4] = {};

  for (int s = 0; s < 5; ++s) {
    const float* A = (s == 0) ? x      : agg    + (size_t)(s - 1) * NN * DD;
    const float* W = (s == 0) ? root_w : conv_w + (size_t)(s - 1) * DD * DD;
    const float* arowp = A + (size_t)arow * DD;
#pragma unroll
    for (int k4 = 0; k4 < 16; ++k4) {
      const int kk = k4 * 4 + koff;
      v2f a = *(const v2f*)(arowp + kk);             // global_load_b64
#pragma unroll
      for (int n = 0; n < 4; ++n) {
        const int col = n * 16 + ncol;
        v2f b = *(const v2f*)(W + (size_t)col * DD + kk);  // W[col][kk..kk+1]
        acc[n] = __builtin_amdgcn_wmma_f32_16x16x4_f32(
            /*neg_a=*/false, a, /*neg_b=*/false, b,
            /*c_mod=*/(short)0, acc[n],
            /*reuse_a=*/false, /*reuse_b=*/false);
      }
    }
  }

  // Add root bias and store the 16x64 tile.
  const int rbase = (lane >> 4) * 8;
#pragma unroll
  for (int n = 0; n < 4; ++n) {
    const float bias = root_b[n * 16 + ncol];
#pragma unroll
    for (int v = 0; v < 8; ++v) {
      out[(size_t)(m0 + rbase + v) * DD + n * 16 + ncol] = acc[n][v] + bias;
    }
  }
}

// ---------------------------------------------------------------------------
extern "C" void kernel_launch(void* const* d_in, const int* in_sizes, int n_in,
                              void* d_out, int out_size, void* d_ws, size_t ws_size,
                              hipStream_t stream) {
  (void)in_sizes; (void)n_in; (void)out_size; (void)ws_size;

  const float* x      = (const float*)d_in[0];
  const float* root_w = (const float*)d_in[1];
  const float* root_b = (const float*)d_in[2];
  const float* conv_w = (const float*)d_in[3];
  const int*   e0     = (const int*)d_in[4];
  const int*   e1     = (const int*)d_in[5];
  const int*   e2     = (const int*)d_in[6];
  const int*   e3     = (const int*)d_in[7];

  float* agg = (float*)d_ws;           // 4 * 100000 * 64 f32 = 97.7 MB
  float* out = (float*)d_out;

  // 0) zero aggregation buffers (must happen every call; atomics accumulate)
  const int n4 = NR * NN * DD / 4;
  zero_ws_kernel<<<2048, 256, 0, stream>>>((float4*)agg, n4);

  // 1) scatter x[src] -> agg_r[dst]
  dim3 sgrid(1024, NR);
  scatter_agg_kernel<<<sgrid, 256, 0, stream>>>(x, e0, e1, e2, e3, agg);

  // 2) fused WMMA GEMM: root + 4 relation matmuls + bias
  const int mtiles = NN / 16;                    // 6250
  const int blocks = (mtiles + 7) / 8;           // 8 waves/block
  rgcn_wmma_gemm_kernel<<<blocks, 256, 0, stream>>>(x, root_w, root_b, conv_w,
                                                    agg, out);
}